// Texture_15831249453087
// MI455X (gfx1250) — compile-verified
//
#include <hip/hip_runtime.h>

// Multi-scale bilinear grid_sample (zeros padding, align_corners=False).
//
// Per pixel, per layer: ix = gx_raw*W - 0.5 (folding the x*2-1 -> ((g+1)*W-1)*0.5
// chain), 2x2 taps. Taps (x0,y) and (x0+1,y) are adjacent floats, so each row
// is fetched with ONE b64 gather from a column base clamped to [0, W-2]; a
// branch-free swap-select recovers the correct value when x0 is clamped, and
// out-of-range taps are handled by zeroing their weights (== zeros padding).
//
// Streaming grid loads / output stores use non-temporal hints so the 201 MB of
// streaming traffic does not evict the ~89 MB texture working set from the
// 192 MB L2 (the gathers are L2-resident after warm-up).

// Native clang vector type: required by __builtin_nontemporal_load (HIP's
// float2 is a class and is rejected), and guarantees b64 lowering for gathers.
typedef float v2f __attribute__((ext_vector_type(2)));

__device__ __forceinline__ float sample_bilinear(const float* __restrict__ tex,
                                                 const int W, const int H,
                                                 const float xr, const float yr) {
    // xr, yr are the raw uniform [0,1] inputs (before the *2-1 remap).
    const float ix = __fmaf_rn(xr, (float)W, -0.5f);
    const float iy = __fmaf_rn(yr, (float)H, -0.5f);
    const float fx0 = floorf(ix);
    const float fy0 = floorf(iy);
    float wx1 = ix - fx0;
    float wy1 = iy - fy0;
    float wx0 = 1.0f - wx1;
    float wy0 = 1.0f - wy1;

    const int x0 = (int)fx0;
    const int y0 = (int)fy0;
    const int x1 = x0 + 1;
    const int y1 = y0 + 1;

    // Zeros padding: kill the weight of any out-of-range tap.
    if (x0 < 0 || x0 >= W) wx0 = 0.0f;
    if (x1 < 0 || x1 >= W) wx1 = 0.0f;
    if (y0 < 0 || y0 >= H) wy0 = 0.0f;
    if (y1 < 0 || y1 >= H) wy1 = 0.0f;

    // Clamped addresses: column pair base in [0, W-2] keeps the 8-byte load
    // in-bounds for every row; rows clamp to [0, H-1].
    const int xp  = min(max(x0, 0), W - 2);
    const int y0c = min(max(y0, 0), H - 1);
    const int y1c = min(max(y1, 0), H - 1);
    const bool at_base = (x0 == xp);

    // 4-byte-aligned b64 gathers (CDNA5 DWORD alignment mode handles this).
    const v2f p0 = *(const v2f*)(tex + (unsigned)(y0c * W + xp));
    const v2f p1 = *(const v2f*)(tex + (unsigned)(y1c * W + xp));

    // Swap-select: if x0 was clamped up (x0==-1) the valid x1 tap is p.x;
    // if clamped down (x0==W-1) the valid x0 tap is p.y. Invalid side has w=0.
    const float v00 = at_base ? p0.x : p0.y;
    const float v10 = at_base ? p0.y : p0.x;
    const float v01 = at_base ? p1.x : p1.y;
    const float v11 = at_base ? p1.y : p1.x;

    return wy0 * (wx0 * v00 + wx1 * v10) + wy1 * (wx0 * v01 + wx1 * v11);
}

__global__ __launch_bounds__(256) void texture_ms_grid_sample_kernel(
    const float* __restrict__ grid,   // N x 2 raw uniform coords
    const float* __restrict__ l1,     // 4096 x 4096
    const float* __restrict__ l2,     // 2048 x 2048
    const float* __restrict__ l3,     // 1024 x 1024
    const float* __restrict__ l4,     //  512 x  512
    float* __restrict__ out,          // N
    const int N) {
    const int idx = blockIdx.x * blockDim.x + threadIdx.x;
    if (idx >= N) return;

    // Prefetch the grid stream a few cachelines ahead (gfx1250 global_prefetch_b8).
    {
        int pf = idx + 4096;
        pf = pf < N ? pf : N - 1;
        __builtin_prefetch(grid + 2 * pf, 0, 0);
    }

    // Non-temporal streaming load of this pixel's (gx, gy).
    const v2f g = __builtin_nontemporal_load((const v2f*)grid + idx);
    const float xr = g.x;   // maps to width  (W) axis
    const float yr = g.y;   // maps to height (H) axis

    float acc;
    acc  = sample_bilinear(l1, 4096, 4096, xr, yr);
    acc += sample_bilinear(l2, 2048, 2048, xr, yr);
    acc += sample_bilinear(l3, 1024, 1024, xr, yr);
    acc += sample_bilinear(l4,  512,  512, xr, yr);

    // Non-temporal store: output is write-once streaming data.
    __builtin_nontemporal_store(acc, out + idx);
}

extern "C" void kernel_launch(void* const* d_in, const int* in_sizes, int n_in,
                              void* d_out, int out_size, void* d_ws, size_t ws_size,
                              hipStream_t stream) {
    const float* x  = (const float*)d_in[0];   // (1, 4096, 4096, 2)
    const float* l1 = (const float*)d_in[1];   // (1, 1, 4096, 4096)
    const float* l2 = (const float*)d_in[2];   // (1, 1, 2048, 2048)
    const float* l3 = (const float*)d_in[3];   // (1, 1, 1024, 1024)
    const float* l4 = (const float*)d_in[4];   // (1, 1,  512,  512)
    float* out = (float*)d_out;                // (1, 1, 4096, 4096)

    const int N = 4096 * 4096;
    const int block = 256;                     // 8 wave32 waves per block
    const int blocks = (N + block - 1) / block;

    texture_ms_grid_sample_kernel<<<blocks, block, 0, stream>>>(
        x, l1, l2, l3, l4, out, N);
}